// TeacherForcer_33629593928300
// MI455X (gfx1250) — compile-verified
//
#include <hip/hip_runtime.h>
#include <math.h>

#define NEGF (-1.0e9f)

typedef float v2f __attribute__((ext_vector_type(2)));
typedef float v8f __attribute__((ext_vector_type(8)));

union V8 { v8f v; float f[8]; };

__device__ inline v8f wmma4(v2f a, v2f b, v8f c) {
  // D = A(16x4 f32) * B(4x16 f32) + C(16x16 f32)
  return __builtin_amdgcn_wmma_f32_16x16x4_f32(false, a, false, b, (short)0, c, false, false);
}

// A fragment: 16x4 f32, row-major source with stride `stride`.
// lanes 0-15 -> M=lane, K={k0,k0+1}; lanes 16-31 -> K={k0+2,k0+3}.
__device__ inline v2f ldA(const float* A, int stride, int m0, int k0, int lane) {
  int m = lane & 15, hi = lane >> 4;
  v2f a;
  a.x = A[(m0 + m) * stride + k0 + hi * 2];
  a.y = A[(m0 + m) * stride + k0 + hi * 2 + 1];
  return a;
}
// B fragment: 4x16 f32, row-major KxN source.
__device__ inline v2f ldB(const float* B, int stride, int k0, int n0, int lane) {
  int c = lane & 15, hi = lane >> 4;
  v2f b;
  b.x = B[(k0 + hi * 2) * stride + n0 + c];
  b.y = B[(k0 + hi * 2 + 1) * stride + n0 + c];
  return b;
}
__device__ inline v2f ldBg(const float* B, int stride, int k0, int n0, int lane, int kvalid) {
  int c = lane & 15, hi = lane >> 4;
  int r0 = k0 + hi * 2, r1 = r0 + 1;
  v2f b;
  b.x = (r0 < kvalid) ? B[r0 * stride + n0 + c] : 0.f;
  b.y = (r1 < kvalid) ? B[r1 * stride + n0 + c] : 0.f;
  return b;
}

// ---------------- generic helpers ----------------
__global__ void k_zero_f32(float* p, int n) {
  int i = blockIdx.x * blockDim.x + threadIdx.x;
  if (i < n) p[i] = 0.f;
}

// ---------------- pocket GCN ----------------
__global__ void k_pdeg(const int* dst, float* deg, int E) {
  int e = blockIdx.x * blockDim.x + threadIdx.x;
  if (e < E) atomicAdd(&deg[dst[e]], 1.f);
}

__global__ void k_pprep(const float* x, const float* deg, float* dinv, float* c,
                        float* agg, int n) {
  int i = blockIdx.x * blockDim.x + threadIdx.x;
  if (i >= n) return;
  float d = rsqrtf(deg[i] + 1.f);  // +1 self loop
  dinv[i] = d;
  c[i] = d * d;  // self-loop src contribution
  float dd = d * d;
  for (int k = 0; k < 15; ++k) agg[i * 16 + k] = dd * x[i * 15 + k];
  agg[i * 16 + 15] = 0.f;
}

__global__ void k_pscatter(const float* x, const int* src, const int* dst,
                           const float* dinv, float* c, float* agg, int E) {
  int e = blockIdx.x * blockDim.x + threadIdx.x;
  if (e >= E) return;
  int s = src[e], d = dst[e];
  float nrm = dinv[s] * dinv[d];
  atomicAdd(&c[s], nrm);
  for (int k = 0; k < 15; ++k) atomicAdd(&agg[d * 16 + k], nrm * x[s * 15 + k]);
}

// per-block: h1_tile = relu(agg_tile @ W1 + b1); colsum += c[row]*h1_tile rows
__global__ __launch_bounds__(256) void k_pocket_gemm(const float* agg, const float* c,
                                                     const float* W1, const float* b1,
                                                     float* part, int n) {
  __shared__ float As[64 * 16];
  __shared__ float Bs[16 * 256];
  __shared__ float cs[64];
  __shared__ float colsum[256];
  int tid = threadIdx.x;
  int base = blockIdx.x * 64;
  for (int idx = tid; idx < 1024; idx += 256) {
    int r = idx >> 4;
    int g = base + r;
    As[idx] = (g < n) ? agg[g * 16 + (idx & 15)] : 0.f;
  }
  for (int idx = tid; idx < 4096; idx += 256) {
    int r = idx >> 8;
    Bs[idx] = (r < 15) ? W1[r * 256 + (idx & 255)] : 0.f;
  }
  if (tid < 64) {
    int g = base + tid;
    cs[tid] = (g < n) ? c[g] : 0.f;
  }
  colsum[tid] = 0.f;
  __syncthreads();
  int wid = tid >> 5, lane = tid & 31;
  for (int t = 0; t < 8; ++t) {
    int tile = wid * 8 + t, mt = tile >> 4, nt = tile & 15;
    V8 acc;
#pragma unroll
    for (int q = 0; q < 8; ++q) acc.f[q] = 0.f;
#pragma unroll
    for (int k0 = 0; k0 < 16; k0 += 4) {
      v2f a = ldA(As, 16, mt * 16, k0, lane);
      v2f b = ldB(Bs, 256, k0, nt * 16, lane);
      acc.v = wmma4(a, b, acc.v);
    }
    int col = nt * 16 + (lane & 15);
    int hi = lane >> 4;
    float bias = b1[col];
#pragma unroll
    for (int q = 0; q < 8; ++q) {
      int m = mt * 16 + q + hi * 8;
      float vv = acc.f[q] + bias;
      vv = vv > 0.f ? vv : 0.f;
      vv *= cs[m];
      atomicAdd(&colsum[col], vv);
    }
  }
  __syncthreads();
  part[blockIdx.x * 256 + tid] = colsum[tid];
}

__global__ void k_vecred(const float* part, float* vec, int nblocks) {
  int n = threadIdx.x;
  float s = 0.f;
  for (int b = 0; b < nblocks; ++b) s += part[b * 256 + n];  // fixed order
  vec[n] = s;
}

__global__ void k_pfinal(const float* vec, const float* W2, const float* b2,
                         float* zp, float n) {
  int j = threadIdx.x;
  float s = 0.f;
  for (int k = 0; k < 256; ++k) s += vec[k] * W2[k * 256 + j];
  zp[j] = s / n + b2[j];
}

// ---------------- ligand GCN ----------------
__global__ __launch_bounds__(256) void k_lig_prep(const float* geom, const int* label,
                                                  const int* srcl, const int* dstl,
                                                  float* xl, float* labv, float* dinvl,
                                                  float* agg1) {
  __shared__ float X[64 * 16];
  __shared__ float dg[64];
  __shared__ float dv[64];
  __shared__ float A1[64 * 16];
  int tid = threadIdx.x;
  for (int idx = tid; idx < 1024; idx += 256) {
    int i = idx >> 4, k = idx & 15;
    float v = 0.f;
    if (i < 48) {
      if (k < 4) v = geom[i * 4 + k];
      else if (k < 15) v = (label[i] == (k - 4)) ? 1.f : 0.f;
    }
    X[idx] = v;
    float lv = 0.f;
    if (k < 11) {
      if (i < 48) lv = (label[i] == k) ? 1.f : 0.f;
      else if (i == 48) lv = (k == 10) ? 1.f : 0.f;
    }
    labv[idx] = lv;
  }
  if (tid < 64) dg[tid] = 1.f;
  __syncthreads();
  for (int e = tid; e < 144; e += 256) atomicAdd(&dg[dstl[e]], 1.f);
  __syncthreads();
  if (tid < 64) {
    dv[tid] = (tid < 48) ? rsqrtf(dg[tid]) : 0.f;
    dinvl[tid] = dv[tid];
  }
  __syncthreads();
  for (int idx = tid; idx < 1024; idx += 256) {
    int i = idx >> 4;
    A1[idx] = dv[i] * dv[i] * X[idx];
  }
  __syncthreads();
  for (int w = tid; w < 144 * 16; w += 256) {
    int e = w >> 4, k = w & 15;
    int s = srcl[e], d = dstl[e];
    atomicAdd(&A1[d * 16 + k], dv[s] * dv[d] * X[s * 16 + k]);
  }
  __syncthreads();
  for (int idx = tid; idx < 1024; idx += 256) {
    agg1[idx] = A1[idx];
    xl[idx] = X[idx];
  }
}

// single-block 64xN GEMM: out = (relu?)(A @ B + bias), A is 64 x lda, B KxN(256)
__global__ __launch_bounds__(256) void k_gemm64(const float* A, int lda, int K,
                                                int kvalid, const float* B,
                                                const float* bias, float* out,
                                                int relu) {
  int tid = threadIdx.x;
  int wid = tid >> 5, lane = tid & 31;
  for (int t = 0; t < 8; ++t) {
    int tile = wid * 8 + t, mt = tile >> 4, nt = tile & 15;
    V8 acc;
#pragma unroll
    for (int q = 0; q < 8; ++q) acc.f[q] = 0.f;
    for (int k0 = 0; k0 < K; k0 += 4) {
      v2f a = ldA(A, lda, mt * 16, k0, lane);
      v2f b = ldBg(B, 256, k0, nt * 16, lane, kvalid);
      acc.v = wmma4(a, b, acc.v);
    }
    int col = nt * 16 + (lane & 15);
    int hi = lane >> 4;
    float bb = bias[col];
#pragma unroll
    for (int q = 0; q < 8; ++q) {
      int m = mt * 16 + q + hi * 8;
      float vv = acc.f[q] + bb;
      if (relu) vv = vv > 0.f ? vv : 0.f;
      out[m * 256 + col] = vv;
    }
  }
}

__global__ __launch_bounds__(256) void k_lig_agg2(const float* h, const float* dinvl,
                                                  const int* srcl, const int* dstl,
                                                  float* a2) {
  int tid = threadIdx.x;
  for (int idx = tid; idx < 64 * 256; idx += 256) {
    int i = idx >> 8;
    a2[idx] = (i < 48) ? dinvl[i] * dinvl[i] * h[idx] : 0.f;
  }
  __threadfence();
  __syncthreads();
  for (int w = tid; w < 144 * 256; w += 256) {
    int e = w >> 8, cc = w & 255;
    int s = srcl[e], d = dstl[e];
    atomicAdd(&a2[d * 256 + cc], dinvl[s] * dinvl[d] * h[s * 256 + cc]);
  }
}

// H_init (267) + ligand label log-prob
__global__ __launch_bounds__(256) void k_lig_head(const float* zl, const int* label,
                                                  const float* Wf, const float* bf,
                                                  float* Hinit, float* label_logp) {
  __shared__ float red[256];
  int tid = threadIdx.x;
  for (int idx = tid; idx < 267; idx += 256) {
    float s = 0.f;
    if (idx < 256) {
      for (int i = 0; i < 48; ++i) s += zl[i * 256 + idx];
    } else {
      int k = idx - 256;
      for (int i = 0; i < 48; ++i) s += (label[i] == k) ? 1.f : 0.f;
    }
    Hinit[idx] = s / 48.f;
  }
  float lp = 0.f;
  if (tid < 48) {
    float lg[11];
    float mx = -3.0e38f;
    for (int a = 0; a < 11; ++a) {
      float s = bf[a] + ((a == 10) ? NEGF : 0.f);
      for (int k = 0; k < 256; ++k) s += zl[tid * 256 + k] * Wf[k * 11 + a];
      lg[a] = s;
      if (s > mx) mx = s;
    }
    float se = 0.f;
    for (int a = 0; a < 11; ++a) se += expf(lg[a] - mx);
    lp = lg[label[tid]] - (mx + logf(se));
  }
  red[tid] = (tid < 48) ? lp : 0.f;
  __syncthreads();
  if (tid == 0) {
    float s = 0.f;
    for (int i = 0; i < 48; ++i) s += red[i];  // fixed order
    *label_logp = s;
  }
}

// ---------------- teacher-forced scan: integer/mask simulation ----------------
__global__ void k_simulate(const float* labv, const int* bfs, const int* bond,
                           int* esrc, int* edst, int* Kcnt, int* su, int* sv, int* sb,
                           int* sact, int* sdoe, int* scnt, float* stime, float* sm,
                           float* sev) {
  if (threadIdx.x != 0 || blockIdx.x != 0) return;
  const float VAL[11] = {4.f, 1.f, 3.f, 1.f, 2.f, 1.f, 5.f, 1.f, 6.f, 1.f, 1000.f};
  const float BV[4] = {1.f, 2.f, 3.f, 1.5f};
  float val[49];
  bool closed[49];
  bool adj[49][49];
  int Q[97];
  for (int i = 0; i < 49; ++i) {
    float s = 0.f;
    for (int k = 0; k < 11; ++k) s += labv[i * 16 + k] * VAL[k];
    val[i] = s;
    closed[i] = false;
    for (int j = 0; j < 49; ++j) adj[i][j] = false;
  }
  int head = 0, tail = 1, k = 0;
  Q[0] = bfs[0];
  for (int t = 0; t < 96; ++t) {
    int v = bfs[t * 2 + 1];
    int b = bond[t];
    int active = (tail > head) ? 1 : 0;
    int u = Q[head];
    bool ulow = (val[u] < 1.f);
    float m_nou[49];
    for (int i = 0; i < 49; ++i) {
      float base = ((val[i] < 1.f) || closed[i] || adj[u][i]) ? NEGF : 0.f;
      if (i == u) base = NEGF;
      m_nou[i] = ulow ? NEGF : base;
    }
    float mnv = m_nou[v];
    for (int i = 0; i < 49; ++i) {
      float mm = m_nou[i];
      if (i == v) mm = 0.f;
      if (i == 48) mm = 0.f;
      sm[t * 64 + i] = mm;
    }
    su[t] = u; sv[t] = v; sb[t] = b; sact[t] = active; scnt[t] = k;
    stime[t] = (float)k;
    int is_stop = (v == 48);
    int pop = active && is_stop;
    int doe = active && !is_stop;
    sdoe[t] = doe;
    if (pop) { closed[u] = true; head++; }
    if (doe) {
      if (tail < 97) Q[tail] = v;
      tail++;
      if (2 * k + 1 < 192) { esrc[2 * k] = u; edst[2 * k] = v; esrc[2 * k + 1] = v; edst[2 * k + 1] = u; }
      k++;
    }
    float mv = fminf(val[u], val[v]);
    float ev[4] = {0.f, 0.f, 0.f, 0.f};
    if (mv <= 2.f) { ev[2] = NEGF; }
    if (mv <= 1.f) { ev[1] = NEGF; ev[2] = NEGF; }
    if (mv <= 0.f) { ev[0] = NEGF; ev[1] = NEGF; ev[2] = NEGF; ev[3] = NEGF; }
    if (mnv <= NEGF * 0.5f) { ev[0] = NEGF; ev[1] = NEGF; ev[2] = NEGF; ev[3] = NEGF; }
    ev[b] = 0.f;
    for (int q = 0; q < 4; ++q) sev[t * 4 + q] = ev[q];
    float dec = doe ? BV[b] : 0.f;
    val[u] -= dec;
    val[v] -= dec;
    if (doe) { adj[u][v] = true; adj[v][u] = true; }
  }
  *Kcnt = k;
}

// ---------------- parallel prefix-graph GCNs (block j -> state j) ----------------
__global__ __launch_bounds__(256) void k_gcn_dyn(const float* labv, const int* esrc,
                                                 const int* edst, const int* Kcnt,
                                                 const int* bfs, const float* Wd1,
                                                 const float* bd1, const float* Wd2,
                                                 const float* bd2, float* hbuf,
                                                 float* a2buf, float* ZV) {
  int j = blockIdx.x;
  int tid = threadIdx.x;
  __shared__ float Ls[64 * 16];
  __shared__ float Bs[16 * 256];
  __shared__ float agg1[64 * 16];
  __shared__ float dinv[64];
  __shared__ float degs[64];
  __shared__ int Es[2 * 192];
  __shared__ int cntS;
  for (int idx = tid; idx < 1024; idx += 256) Ls[idx] = labv[idx];
  if (tid == 0) {
    int cnt;
    if (j == 0) {
      cnt = 1;
      Es[0] = bfs[0];
      Es[192] = bfs[1];
    } else {
      int K = *Kcnt;
      cnt = 2 * j;
      if (cnt > 2 * K) cnt = 2 * K;
    }
    cntS = cnt;
  }
  __syncthreads();
  int cnt = cntS;
  if (j != 0) {
    for (int e = tid; e < cnt; e += 256) { Es[e] = esrc[e]; Es[192 + e] = edst[e]; }
  }
  if (tid < 64) degs[tid] = 1.f;  // self loops
  __syncthreads();
  for (int e = tid; e < cnt; e += 256) atomicAdd(&degs[Es[192 + e]], 1.f);
  __syncthreads();
  if (tid < 64) dinv[tid] = (tid < 49) ? rsqrtf(degs[tid]) : 0.f;
  __syncthreads();
  for (int idx = tid; idx < 1024; idx += 256) {
    int i = idx >> 4;
    agg1[idx] = dinv[i] * dinv[i] * Ls[idx];
  }
  __syncthreads();
  for (int w = tid; w < cnt * 16; w += 256) {
    int e = w >> 4, kk = w & 15;
    int s = Es[e], d = Es[192 + e];
    atomicAdd(&agg1[d * 16 + kk], dinv[s] * dinv[d] * Ls[s * 16 + kk]);
  }
  for (int idx = tid; idx < 4096; idx += 256) {
    int r = idx >> 8;
    Bs[idx] = (r < 11) ? Wd1[r * 256 + (idx & 255)] : 0.f;
  }
  __syncthreads();
  float* h = hbuf + (size_t)j * 64 * 256;
  int wid = tid >> 5, lane = tid & 31;
  for (int t = 0; t < 8; ++t) {
    int tile = wid * 8 + t, mt = tile >> 4, nt = tile & 15;
    V8 acc;
#pragma unroll
    for (int q = 0; q < 8; ++q) acc.f[q] = 0.f;
#pragma unroll
    for (int k0 = 0; k0 < 16; k0 += 4) {
      v2f a = ldA(agg1, 16, mt * 16, k0, lane);
      v2f b = ldB(Bs, 256, k0, nt * 16, lane);
      acc.v = wmma4(a, b, acc.v);
    }
    int col = nt * 16 + (lane & 15);
    int hi = lane >> 4;
    float bias = bd1[col];
#pragma unroll
    for (int q = 0; q < 8; ++q) {
      int m = mt * 16 + q + hi * 8;
      float vv = acc.f[q] + bias;
      h[m * 256 + col] = vv > 0.f ? vv : 0.f;
    }
  }
  __threadfence();
  __syncthreads();
  float* a2 = a2buf + (size_t)j * 64 * 256;
  for (int idx = tid; idx < 64 * 256; idx += 256) {
    int i = idx >> 8;
    a2[idx] = (i < 49) ? dinv[i] * dinv[i] * h[idx] : 0.f;
  }
  __threadfence();
  __syncthreads();
  for (int w = tid; w < cnt * 256; w += 256) {
    int e = w >> 8, cc = w & 255;
    int s = Es[e], d = Es[192 + e];
    atomicAdd(&a2[d * 256 + cc], dinv[s] * dinv[d] * h[s * 256 + cc]);
  }
  __threadfence();
  __syncthreads();
  float* z = ZV + (size_t)j * 64 * 256;
  for (int t = 0; t < 8; ++t) {
    int tile = wid * 8 + t, mt = tile >> 4, nt = tile & 15;
    V8 acc;
#pragma unroll
    for (int q = 0; q < 8; ++q) acc.f[q] = 0.f;
    for (int k0 = 0; k0 < 256; k0 += 4) {
      v2f a = ldA(a2, 256, mt * 16, k0, lane);
      v2f b = ldB(Wd2, 256, k0, nt * 16, lane);
      acc.v = wmma4(a, b, acc.v);
    }
    int col = nt * 16 + (lane & 15);
    int hi = lane >> 4;
    float bias = bd2[col];
#pragma unroll
    for (int q = 0; q < 8; ++q) {
      int m = mt * 16 + q + hi * 8;
      z[m * 256 + col] = acc.f[q] + bias;
    }
  }
}

// ---------------- projections of the 97 GCN states onto Wg/Wh segments ----------------
__global__ __launch_bounds__(256) void k_reduce_zv(const float* ZV, const float* Wg,
                                                   const float* Wh, float* Hz, float* gZ,
                                                   float* gU, float* hZ, float* hU,
                                                   float* gH, float* hH) {
  int j = blockIdx.x, tid = threadIdx.x;
  const float* z = ZV + (size_t)j * 64 * 256;
  __shared__ float hz[256];
  float s = 0.f;
  for (int i = 0; i < 49; ++i) s += z[i * 256 + tid];
  s *= (1.f / 49.f);
  hz[tid] = s;
  Hz[j * 256 + tid] = s;
  __syncthreads();
  for (int idx = tid; idx < 495; idx += 256) {
    if (idx < 490) {
      int i = idx / 10, w = idx % 10;
      const float* row = z + i * 256;
      float acc = 0.f;
      if (w == 0) {
        for (int c = 0; c < 256; ++c) acc += Wg[524 + c] * row[c];
        gZ[j * 64 + i] = acc;
      } else if (w == 1) {
        for (int c = 0; c < 256; ++c) acc += Wg[257 + c] * row[c];
        gU[j * 64 + i] = acc;
      } else if (w < 6) {
        int q = w - 2;
        for (int c = 0; c < 256; ++c) acc += Wh[(524 + c) * 4 + q] * row[c];
        hZ[(j * 64 + i) * 4 + q] = acc;
      } else {
        int q = w - 6;
        for (int c = 0; c < 256; ++c) acc += Wh[(257 + c) * 4 + q] * row[c];
        hU[(j * 64 + i) * 4 + q] = acc;
      }
    } else {
      int w = idx - 490;
      float acc = 0.f;
      if (w == 0) {
        for (int c = 0; c < 256; ++c) acc += Wg[791 + c] * hz[c];
        gH[j] = acc;
      } else {
        int q = w - 1;
        for (int c = 0; c < 256; ++c) acc += Wh[(791 + c) * 4 + q] * hz[c];
        hH[j * 4 + q] = acc;
      }
    }
  }
}

// cst layout: [0]=gP [1]=gI [2]=gHlab ; [16+i]=glabrow ; [96+i]=glabu ;
// [160+q]=hP ; [168+q]=hI ; [176+q]=hHlab ; [192+i*4+q]=hLu ; [448+i*4+q]=hLr
__global__ __launch_bounds__(256) void k_consts(const float* labv, const float* zp,
                                                const float* Hinit, const float* Wg,
                                                const float* Wh, float* cst) {
  __shared__ float ml[16];
  int tid = threadIdx.x;
  if (tid < 16) {
    float s = 0.f;
    for (int i = 0; i < 49; ++i) s += labv[i * 16 + tid];
    ml[tid] = s / 49.f;
  }
  __syncthreads();
  if (tid == 0) {
    float s = 0.f;
    for (int c = 0; c < 256; ++c) s += Wg[1 + c] * zp[c];
    cst[0] = s;
  }
  if (tid == 1) {
    float s = 0.f;
    for (int c = 0; c < 267; ++c) s += Wg[1058 + c] * Hinit[c];
    cst[1] = s;
  }
  if (tid == 2) {
    float s = 0.f;
    for (int c = 0; c < 11; ++c) s += Wg[1047 + c] * ml[c];
    cst[2] = s;
  }
  if (tid >= 16 && tid < 65) {
    int i = tid - 16;
    float s = 0.f;
    for (int c = 0; c < 11; ++c) s += Wg[780 + c] * labv[i * 16 + c];
    cst[16 + i] = s;
  }
  if (tid >= 96 && tid < 145) {
    int i = tid - 96;
    float s = 0.f;
    for (int c = 0; c < 11; ++c) s += Wg[513 + c] * labv[i * 16 + c];
    cst[96 + i] = s;
  }
  if (tid >= 160 && tid < 164) {
    int q = tid - 160;
    float s = 0.f;
    for (int c = 0; c < 256; ++c) s += Wh[(1 + c) * 4 + q] * zp[c];
    cst[160 + q] = s;
  }
  if (tid >= 168 && tid < 172) {
    int q = tid - 168;
    float s = 0.f;
    for (int c = 0; c < 267; ++c) s += Wh[(1058 + c) * 4 + q] * Hinit[c];
    cst[168 + q] = s;
  }
  if (tid >= 176 && tid < 180) {
    int q = tid - 176;
    float s = 0.f;
    for (int c = 0; c < 11; ++c) s += Wh[(1047 + c) * 4 + q] * ml[c];
    cst[176 + q] = s;
  }
  for (int idx = tid; idx < 49 * 4 * 2; idx += 256) {
    int half = idx / 196;
    int r = idx % 196;
    int i = r / 4, q = r % 4;
    int base = half ? 780 : 513;
    float s = 0.f;
    for (int c = 0; c < 11; ++c) s += Wh[(base + c) * 4 + q] * labv[i * 16 + c];
    cst[(half ? 448 : 192) + i * 4 + q] = s;
  }
}

__global__ __launch_bounds__(64) void k_step(const int* su, const int* sv, const int* sb,
                                             const int* sact, const int* sdoe,
                                             const int* scnt, const float* stime,
                                             const float* sm, const float* sev,
                                             const float* gZ, const float* gU,
                                             const float* hZ, const float* hU,
                                             const float* gH, const float* hH,
                                             const float* cst, const float* bg,
                                             const float* bh, const float* Wg,
                                             const float* Wh, float* contrib) {
  int t = blockIdx.x, tid = threadIdx.x;
  __shared__ float red[64];
  int active = sact[t];
  if (!active) {
    if (tid == 0) contrib[t] = 0.f;
    return;
  }
  int u = su[t], v = sv[t], b = sb[t], g = scnt[t], doe = sdoe[t];
  float time = stime[t];
  float gconst = bg[0] + Wg[0] * time + cst[0] + gU[g * 64 + u] + cst[96 + u] + gH[g] +
                 cst[2] + cst[1];
  float gl = -3.0e38f;
  if (tid < 49) gl = gconst + gZ[g * 64 + tid] + cst[16 + tid] + sm[t * 64 + tid];
  red[tid] = gl;
  __syncthreads();
  for (int o = 32; o > 0; o >>= 1) {
    if (tid < o) {
      float a = red[tid], bb = red[tid + o];
      red[tid] = a > bb ? a : bb;
    }
    __syncthreads();
  }
  float mx = red[0];
  __syncthreads();
  red[tid] = (tid < 49) ? expf(gl - mx) : 0.f;
  __syncthreads();
  for (int o = 32; o > 0; o >>= 1) {
    if (tid < o) red[tid] += red[tid + o];
    __syncthreads();
  }
  if (tid == 0) {
    float lse = mx + logf(red[0]);
    float glv = gconst + gZ[g * 64 + v] + cst[16 + v] + sm[t * 64 + v];
    float c = glv - lse;
    if (doe) {
      float hl[4];
      float hmx = -3.0e38f;
      for (int q = 0; q < 4; ++q) {
        float s = bh[q] + Wh[q] * time + cst[160 + q] + hU[(g * 64 + u) * 4 + q] +
                  cst[192 + u * 4 + q] + hZ[(g * 64 + v) * 4 + q] + cst[448 + v * 4 + q] +
                  hH[g * 4 + q] + cst[176 + q] + cst[168 + q] + sev[t * 4 + q];
        hl[q] = s;
        if (s > hmx) hmx = s;
      }
      float se = 0.f;
      for (int q = 0; q < 4; ++q) se += expf(hl[q] - hmx);
      c += hl[b] - (hmx + logf(se));
    }
    contrib[t] = c;
  }
}

__global__ void k_final(const float* contrib, const float* label_logp, float* out) {
  if (threadIdx.x == 0 && blockIdx.x == 0) {
    float s = *label_logp;
    for (int t = 0; t < 96; ++t) s += contrib[t];  // fixed order
    out[0] = s;
  }
}

// ---------------- host launch ----------------
extern "C" void kernel_launch(void* const* d_in, const int* in_sizes, int n_in,
                              void* d_out, int out_size, void* d_ws, size_t ws_size,
                              hipStream_t stream) {
  (void)in_sizes; (void)n_in; (void)out_size; (void)ws_size;
  const float* x_p = (const float*)d_in[0];
  const int* src_p = (const int*)d_in[1];
  const int* dst_p = (const int*)d_in[2];
  const float* x_l_geom = (const float*)d_in[3];
  const int* x_l_label = (const int*)d_in[4];
  const int* src_l = (const int*)d_in[5];
  const int* dst_l = (const int*)d_in[6];
  const int* bfs = (const int*)d_in[7];
  const int* bond = (const int*)d_in[8];
  const float* Wp1 = (const float*)d_in[9];
  const float* bp1 = (const float*)d_in[10];
  const float* Wp2 = (const float*)d_in[11];
  const float* bp2 = (const float*)d_in[12];
  const float* Wl1 = (const float*)d_in[13];
  const float* bl1 = (const float*)d_in[14];
  const float* Wl2 = (const float*)d_in[15];
  const float* bl2 = (const float*)d_in[16];
  const float* Wd1 = (const float*)d_in[17];
  const float* bd1 = (const float*)d_in[18];
  const float* Wd2 = (const float*)d_in[19];
  const float* bd2 = (const float*)d_in[20];
  const float* Wf = (const float*)d_in[21];
  const float* bf = (const float*)d_in[22];
  const float* Wg = (const float*)d_in[23];
  const float* bg = (const float*)d_in[24];
  const float* Wh = (const float*)d_in[25];
  const float* bh = (const float*)d_in[26];
  float* out = (float*)d_out;

  const int NP = 30000, EP = 480000;
  const int PB = (NP + 63) / 64;  // pocket gemm blocks = 469

  char* W = (char*)d_ws;
  size_t off = 0;
  auto take = [&](size_t bytes) -> void* {
    off = (off + 255) & ~(size_t)255;
    void* p = W + off;
    off += bytes;
    return p;
  };
  float* deg_p = (float*)take(NP * 4);
  float* dinv_p = (float*)take(NP * 4);
  float* agg_p = (float*)take((size_t)NP * 16 * 4);
  float* c_p = (float*)take(NP * 4);
  float* part = (float*)take((size_t)PB * 256 * 4);
  float* vec = (float*)take(256 * 4);
  float* zp = (float*)take(256 * 4);
  float* xl = (float*)take(64 * 16 * 4);
  float* labv = (float*)take(64 * 16 * 4);
  float* dinvl = (float*)take(64 * 4);
  float* agg1l = (float*)take(64 * 16 * 4);
  float* hl = (float*)take(64 * 256 * 4);
  float* a2l = (float*)take(64 * 256 * 4);
  float* zl = (float*)take(64 * 256 * 4);
  float* Hinit = (float*)take(272 * 4);
  float* label_lp = (float*)take(4);
  int* esrc = (int*)take(192 * 4);
  int* edst = (int*)take(192 * 4);
  int* Kcnt = (int*)take(4);
  int* su = (int*)take(96 * 4);
  int* sv = (int*)take(96 * 4);
  int* sb = (int*)take(96 * 4);
  int* sact = (int*)take(96 * 4);
  int* sdoe = (int*)take(96 * 4);
  int* scnt = (int*)take(96 * 4);
  float* stime = (float*)take(96 * 4);
  float* sm = (float*)take(96 * 64 * 4);
  float* sev = (float*)take(96 * 4 * 4);
  float* ZV = (float*)take((size_t)97 * 64 * 256 * 4);
  float* hdyn = (float*)take((size_t)97 * 64 * 256 * 4);
  float* a2dyn = (float*)take((size_t)97 * 64 * 256 * 4);
  float* Hz = (float*)take(97 * 256 * 4);
  float* gZ = (float*)take(97 * 64 * 4);
  float* gU = (float*)take(97 * 64 * 4);
  float* hZ = (float*)take(97 * 64 * 4 * 4);
  float* hU = (float*)take(97 * 64 * 4 * 4);
  float* gH = (float*)take(97 * 4);
  float* hH = (float*)take(97 * 4 * 4);
  float* cst = (float*)take(1024 * 4);
  float* contrib = (float*)take(96 * 4);

  // ---- pocket GCN (algebraically reduced to aggregation + 2 GEMMs) ----
  k_zero_f32<<<(NP + 255) / 256, 256, 0, stream>>>(deg_p, NP);
  k_pdeg<<<(EP + 255) / 256, 256, 0, stream>>>(dst_p, deg_p, EP);
  k_pprep<<<(NP + 255) / 256, 256, 0, stream>>>(x_p, deg_p, dinv_p, c_p, agg_p, NP);
  k_pscatter<<<(EP + 255) / 256, 256, 0, stream>>>(x_p, src_p, dst_p, dinv_p, c_p, agg_p, EP);
  k_pocket_gemm<<<PB, 256, 0, stream>>>(agg_p, c_p, Wp1, bp1, part, NP);
  k_vecred<<<1, 256, 0, stream>>>(part, vec, PB);
  k_pfinal<<<1, 256, 0, stream>>>(vec, Wp2, bp2, zp, (float)NP);

  // ---- ligand GCN (full 48x256 output) ----
  k_lig_prep<<<1, 256, 0, stream>>>(x_l_geom, x_l_label, src_l, dst_l, xl, labv, dinvl, agg1l);
  k_gemm64<<<1, 256, 0, stream>>>(agg1l, 16, 16, 15, Wl1, bl1, hl, 1);
  k_lig_agg2<<<1, 256, 0, stream>>>(hl, dinvl, src_l, dst_l, a2l);
  k_gemm64<<<1, 256, 0, stream>>>(a2l, 256, 256, 256, Wl2, bl2, zl, 0);
  k_lig_head<<<1, 256, 0, stream>>>(zl, x_l_label, Wf, bf, Hinit, label_lp);

  // ---- teacher-forced scan: deterministic integer/mask simulation ----
  k_simulate<<<1, 32, 0, stream>>>(labv, bfs, bond, esrc, edst, Kcnt, su, sv, sb, sact,
                                   sdoe, scnt, stime, sm, sev);

  // ---- 97 independent prefix-graph GCN states, fully parallel ----
  k_gcn_dyn<<<97, 256, 0, stream>>>(labv, esrc, edst, Kcnt, bfs, Wd1, bd1, Wd2, bd2,
                                    hdyn, a2dyn, ZV);
  k_reduce_zv<<<97, 256, 0, stream>>>(ZV, Wg, Wh, Hz, gZ, gU, hZ, hU, gH, hH);
  k_consts<<<1, 256, 0, stream>>>(labv, zp, Hinit, Wg, Wh, cst);

  // ---- per-step log-prob contributions + deterministic final sum ----
  k_step<<<96, 64, 0, stream>>>(su, sv, sb, sact, sdoe, scnt, stime, sm, sev, gZ, gU,
                                hZ, hU, gH, hH, cst, bg, bh, Wg, Wh, contrib);
  k_final<<<1, 32, 0, stream>>>(contrib, label_lp, out);
}